// GraphAttentionLayer_73778948211304
// MI455X (gfx1250) — compile-verified
//
#include <hip/hip_runtime.h>

typedef __attribute__((ext_vector_type(2))) float v2f;
typedef __attribute__((ext_vector_type(8))) float v8f;

#define ALPHA   0.2f
#define NEG_INF -9000000000000000.0f

// Problem dims (fixed by the reference)
#define BB   8
#define NN   2048
#define FIN  128
#define FOUT 64
#define ROWS (BB * NN)          // 16384 flattened rows
#define TILES (ROWS / 16)       // 1024 16-row tiles
#define KSPLIT 4                // key-range splits per tile
#define KEYS_PER_SPLIT (NN / KSPLIT)   // 512
#define KB 64                   // keys per inner iteration
#define NIT (KEYS_PER_SPLIT / KB)      // 8

// ---------------------------------------------------------------------------
// Kernel 1: Wh = h @ W  (16384x128 @ 128x64), plus f1 = Wh@a1, f2 = Wh@a2.
// One wave per 16-row tile; fp32 WMMA 16x16x4, K-loop of 32 steps.
// ---------------------------------------------------------------------------
__global__ __launch_bounds__(256) void gat_proj(
    const float* __restrict__ h,    // (ROWS, FIN)
    const float* __restrict__ W,    // (FIN, FOUT)
    const float* __restrict__ a,    // (2*FOUT)
    float* __restrict__ Wh,         // (ROWS, FOUT)
    float* __restrict__ f1,         // (ROWS)
    float* __restrict__ f2)         // (ROWS)
{
    const int lane = threadIdx.x & 31;
    const int wv   = threadIdx.x >> 5;
    const int tile = blockIdx.x * 8 + wv;      // 0..1023
    const int r0   = tile * 16;
    const int mrow = lane & 15;                // 0..15
    const int half = lane >> 4;                // 0 or 1

    v8f acc[4];
    #pragma unroll
    for (int t = 0; t < 4; ++t) acc[t] = (v8f)0.0f;

    const float* hrow = h + (size_t)(r0 + mrow) * FIN;

    #pragma unroll 8
    for (int kk = 0; kk < FIN / 4; ++kk) {
        const int kb = kk * 4;
        // A fragment (16x4 f32): VGPR0 = K {kb+0 | kb+2}, VGPR1 = K {kb+1 | kb+3}
        const v2f av = *(const v2f*)(hrow + kb + 2 * half);
        #pragma unroll
        for (int t = 0; t < 4; ++t) {
            // B fragment (4x16 f32): rows striped {K+0|K+2} / {K+1|K+3} over halves
            v2f bv;
            bv.x = W[(kb + 2 * half + 0) * FOUT + t * 16 + mrow];
            bv.y = W[(kb + 2 * half + 1) * FOUT + t * 16 + mrow];
            acc[t] = __builtin_amdgcn_wmma_f32_16x16x4_f32(
                false, av, false, bv, (short)0, acc[t], false, false);
        }
    }

    // Write Wh (C-layout: row m = v + 8*half, col = 16t + mrow)
    #pragma unroll
    for (int t = 0; t < 4; ++t)
        #pragma unroll
        for (int v = 0; v < 8; ++v)
            Wh[(size_t)(r0 + v + 8 * half) * FOUT + t * 16 + mrow] = acc[t][v];

    // f1 = Wh @ a1, f2 = Wh @ a2 : per-lane partials then 16-lane row reduce
    float p1[8], p2[8];
    #pragma unroll
    for (int v = 0; v < 8; ++v) { p1[v] = 0.0f; p2[v] = 0.0f; }
    #pragma unroll
    for (int t = 0; t < 4; ++t) {
        const float a1v = a[t * 16 + mrow];
        const float a2v = a[FOUT + t * 16 + mrow];
        #pragma unroll
        for (int v = 0; v < 8; ++v) {
            p1[v] += acc[t][v] * a1v;
            p2[v] += acc[t][v] * a2v;
        }
    }
    #pragma unroll
    for (int v = 0; v < 8; ++v) {
        float s1 = p1[v], s2 = p2[v];
        #pragma unroll
        for (int d = 1; d < 16; d <<= 1) {
            s1 += __shfl_xor(s1, d, 32);
            s2 += __shfl_xor(s2, d, 32);
        }
        if (mrow == 0) {
            f1[r0 + v + 8 * half] = s1;
            f2[r0 + v + 8 * half] = s2;
        }
    }
}

// ---------------------------------------------------------------------------
// Kernel 2: fused masked-LeakyReLU attention with online softmax + P@Wh.
// One wave = one 16-row query tile x one 512-key split. Produces partial
// (acc 16x64, rowmax M, rowsum S) for later combining.
// ---------------------------------------------------------------------------
__global__ __launch_bounds__(256) void gat_attn(
    const int*   __restrict__ adj,   // (BB, NN, NN)
    const float* __restrict__ Wh,    // (ROWS, FOUT)
    const float* __restrict__ f1,    // (ROWS)
    const float* __restrict__ f2,    // (ROWS)
    float* __restrict__ pacc,        // (TILES*KSPLIT, 16, 64)
    float* __restrict__ pM,          // (TILES*KSPLIT, 16)
    float* __restrict__ pS)          // (TILES*KSPLIT, 16)
{
    __shared__ float ptile[8][16][68];   // per-wave 16x64 P tile, padded

    const int lane  = threadIdx.x & 31;
    const int wv    = threadIdx.x >> 5;
    const int gw    = blockIdx.x * 8 + wv;     // 0..4095
    const int tile  = gw >> 2;                 // 0..1023
    const int split = gw & 3;
    const int b     = tile >> 7;               // 128 tiles per batch
    const int m0    = (tile & 127) * 16;       // query row offset in batch
    const int mrow  = lane & 15;
    const int half  = lane >> 4;
    const int kbeg  = split * KEYS_PER_SPLIT;

    const int*   adjb = adj + (size_t)b * NN * NN + (size_t)m0 * NN;
    const float* Whb  = Wh  + (size_t)b * NN * FOUT;
    const float* f2b  = f2  + (size_t)b * NN;

    v8f acc[4];
    #pragma unroll
    for (int t = 0; t < 4; ++t) acc[t] = (v8f)0.0f;

    float M[8], S[8], F1[8];
    #pragma unroll
    for (int v = 0; v < 8; ++v) {
        M[v] = NEG_INF;
        S[v] = 0.0f;
        F1[v] = f1[tile * 16 + v + 8 * half];
    }

    // Prologue: prefetch first key block (adj as int4, f2 as float4)
    int4   ajc[8];
    float4 f2c;
    #pragma unroll
    for (int v = 0; v < 8; ++v) {
        const int m = v + 8 * half;
        ajc[v] = ((const int4*)(adjb + (size_t)m * NN))[(kbeg >> 2) + mrow];
    }
    f2c = ((const float4*)(f2b + kbeg))[mrow];

    for (int it = 0; it < NIT; ++it) {
        const int k0 = kbeg + it * KB;
        int4   ajn[8];
        float4 f2n;
        const bool more = (it + 1) < NIT;
        if (more) {
            const int k1 = k0 + KB;
            #pragma unroll
            for (int v = 0; v < 8; ++v) {
                const int m = v + 8 * half;
                ajn[v] = ((const int4*)(adjb + (size_t)m * NN))[(k1 >> 2) + mrow];
            }
            f2n = ((const float4*)(f2b + k1))[mrow];
        }

        // --- e / masked LeakyReLU / online softmax, per accumulator row ---
        #pragma unroll
        for (int v = 0; v < 8; ++v) {
            const int m = v + 8 * half;
            const int4 aj = ajc[v];
            float e0 = F1[v] + f2c.x;
            float e1 = F1[v] + f2c.y;
            float e2 = F1[v] + f2c.z;
            float e3 = F1[v] + f2c.w;
            e0 = e0 > 0.0f ? e0 : ALPHA * e0;
            e1 = e1 > 0.0f ? e1 : ALPHA * e1;
            e2 = e2 > 0.0f ? e2 : ALPHA * e2;
            e3 = e3 > 0.0f ? e3 : ALPHA * e3;
            e0 = aj.x > 0 ? e0 : NEG_INF;
            e1 = aj.y > 0 ? e1 : NEG_INF;
            e2 = aj.z > 0 ? e2 : NEG_INF;
            e3 = aj.w > 0 ? e3 : NEG_INF;

            float bm = fmaxf(fmaxf(e0, e1), fmaxf(e2, e3));
            #pragma unroll
            for (int d = 1; d < 16; d <<= 1) bm = fmaxf(bm, __shfl_xor(bm, d, 32));

            const float mn = fmaxf(M[v], bm);
            const float sc = __expf(M[v] - mn);
            const float p0 = __expf(e0 - mn);
            const float p1 = __expf(e1 - mn);
            const float p2 = __expf(e2 - mn);
            const float p3 = __expf(e3 - mn);
            float ps = (p0 + p1) + (p2 + p3);
            #pragma unroll
            for (int d = 1; d < 16; d <<= 1) ps += __shfl_xor(ps, d, 32);

            S[v] = S[v] * sc + ps;
            M[v] = mn;
            #pragma unroll
            for (int t = 0; t < 4; ++t) acc[t][v] = acc[t][v] * sc;

            float4 pv; pv.x = p0; pv.y = p1; pv.z = p2; pv.w = p3;
            *(float4*)&ptile[wv][m][4 * mrow] = pv;   // 16B-aligned LDS store
        }

        // Ensure the P tile is in LDS before reading it back in A-layout.
        asm volatile("s_wait_dscnt 0" ::: "memory");

        // --- acc += P(16xKB) @ Wh(KBx64) via fp32 WMMA, K-step 4 ---
        #pragma unroll
        for (int kk = 0; kk < KB / 4; ++kk) {
            const int kb = kk * 4;
            const v2f av = *(const v2f*)&ptile[wv][mrow][kb + 2 * half];
            #pragma unroll
            for (int t = 0; t < 4; ++t) {
                v2f bv;
                bv.x = Whb[(size_t)(k0 + kb + 2 * half + 0) * FOUT + t * 16 + mrow];
                bv.y = Whb[(size_t)(k0 + kb + 2 * half + 1) * FOUT + t * 16 + mrow];
                acc[t] = __builtin_amdgcn_wmma_f32_16x16x4_f32(
                    false, av, false, bv, (short)0, acc[t], false, false);
            }
        }

        // rotate prefetch
        if (more) {
            #pragma unroll
            for (int v = 0; v < 8; ++v) ajc[v] = ajn[v];
            f2c = f2n;
        }
    }

    // Store partial results for the combine pass.
    #pragma unroll
    for (int t = 0; t < 4; ++t)
        #pragma unroll
        for (int v = 0; v < 8; ++v)
            pacc[((size_t)gw * 16 + (v + 8 * half)) * FOUT + t * 16 + mrow] = acc[t][v];
    if (mrow == 0) {
        #pragma unroll
        for (int v = 0; v < 8; ++v) {
            pM[gw * 16 + v + 8 * half] = M[v];
            pS[gw * 16 + v + 8 * half] = S[v];
        }
    }
}

// ---------------------------------------------------------------------------
// Kernel 3: combine the KSPLIT partial softmaxes, normalize, ELU, write out.
// ---------------------------------------------------------------------------
__global__ __launch_bounds__(256) void gat_combine(
    const float* __restrict__ pacc,
    const float* __restrict__ pM,
    const float* __restrict__ pS,
    float* __restrict__ out)     // (ROWS, FOUT)
{
    const int idx = blockIdx.x * 256 + threadIdx.x;   // 0..ROWS*FOUT-1
    const int c = idx & (FOUT - 1);
    const int r = idx >> 6;
    const int tile = r >> 4;
    const int m = r & 15;

    float Ms[KSPLIT];
    float Mg = NEG_INF;
    #pragma unroll
    for (int s = 0; s < KSPLIT; ++s) {
        Ms[s] = pM[(tile * KSPLIT + s) * 16 + m];
        Mg = fmaxf(Mg, Ms[s]);
    }
    float Ssum = 0.0f, val = 0.0f;
    #pragma unroll
    for (int s = 0; s < KSPLIT; ++s) {
        const float w = __expf(Ms[s] - Mg);
        Ssum += w * pS[(tile * KSPLIT + s) * 16 + m];
        val  += w * pacc[((size_t)(tile * KSPLIT + s) * 16 + m) * FOUT + c];
    }
    float x = val / Ssum;
    out[idx] = x > 0.0f ? x : (__expf(x) - 1.0f);
}

// ---------------------------------------------------------------------------
extern "C" void kernel_launch(void* const* d_in, const int* in_sizes, int n_in,
                              void* d_out, int out_size, void* d_ws, size_t ws_size,
                              hipStream_t stream) {
    const float* h   = (const float*)d_in[0];   // (8,2048,128) f32
    const int*   adj = (const int*)  d_in[1];   // (8,2048,2048) i32
    const float* W   = (const float*)d_in[2];   // (128,64) f32
    const float* a   = (const float*)d_in[3];   // (128,1) f32
    float* out = (float*)d_out;                 // (8,2048,64) f32

    char* ws = (char*)d_ws;
    float* Wh   = (float*)(ws + 0);                       // 16384*64*4   = 4 MiB
    float* f1   = (float*)(ws + 4194304);                 // 16384*4      = 64 KiB
    float* f2   = (float*)(ws + 4259840);                 // 64 KiB
    float* pacc = (float*)(ws + 4325376);                 // 4096*16*64*4 = 16 MiB
    float* pM   = (float*)(ws + 21102592);                // 4096*16*4    = 256 KiB
    float* pS   = (float*)(ws + 21364736);                // 256 KiB

    gat_proj   <<<TILES / 8,          256, 0, stream>>>(h, W, a, Wh, f1, f2);
    gat_attn   <<<TILES * KSPLIT / 8, 256, 0, stream>>>(adj, Wh, f1, f2, pacc, pM, pS);
    gat_combine<<<ROWS * FOUT / 256,  256, 0, stream>>>(pacc, pM, pS, out);
}